// ProjectRenderScoreModel_76459007803854
// MI455X (gfx1250) — compile-verified
//
#include <hip/hip_runtime.h>
#include <hip/hip_bf16.h>

typedef __attribute__((ext_vector_type(2))) float v2f;
typedef __attribute__((ext_vector_type(8))) float v8f;

#define IMG 256
#define NPTS 128
#define NB3 6
#define PIX (IMG * IMG)              // 65536 pixels per camera image
#define MASK_ELEMS (NB3 * PIX)       // 393216 floats of masks before blobs
#define TILES_PER_CAM (PIX / 16)     // 4096 16-pixel tiles per camera

// ---------------------------------------------------------------------------
// Kernel 1: fold camera scale factors into per-(camera,point) parameters.
// W layout (6 arrays of NB3*NPTS=768 floats):
//   [0]=px  [1]=py  [2]=px^2+py^2  [3]=1/(2*sig^2)  [4]=exponent  [5]=intensity
// ---------------------------------------------------------------------------
__global__ __launch_bounds__(256) void prep_kernel(
    const float* __restrict__ points,        // (2,3,128,2)
    const float* __restrict__ sigmas,        // (2,128)
    const float* __restrict__ exponents,     // (2,128)
    const float* __restrict__ intensities,   // (2,128)
    const float* __restrict__ cam_sig,       // (2,3)
    const float* __restrict__ cam_exp,       // (2,3)
    const float* __restrict__ cam_int,       // (2,3)
    float* __restrict__ W)
{
    int g = blockIdx.x * 256 + threadIdx.x;
    if (g >= NB3 * NPTS) return;
    int b3 = g >> 7;          // 0..5
    int n  = g & (NPTS - 1);  // 0..127
    int b  = b3 / 3;
    int c  = b3 % 3;

    // _normalise_scale_factor: v /= mean; sf = 0.2/max|v-1|; if sf<1: (v-1)*sf+1
    auto norm3 = [&](const float* v) -> float {
        float v0 = v[b * 3 + 0], v1 = v[b * 3 + 1], v2 = v[b * 3 + 2];
        float mean = (v0 + v1 + v2) / 3.0f;
        v0 /= mean; v1 /= mean; v2 /= mean;
        float mx = fmaxf(fabsf(v0 - 1.0f), fmaxf(fabsf(v1 - 1.0f), fabsf(v2 - 1.0f)));
        float sf = 0.2f / mx;
        float w = (c == 0) ? v0 : ((c == 1) ? v1 : v2);
        return (sf < 1.0f) ? ((w - 1.0f) * sf + 1.0f) : w;
    };

    float csn = norm3(cam_sig);
    float cen = norm3(cam_exp);
    float cin = norm3(cam_int);

    float sig = sigmas[b * NPTS + n] * csn;
    float e   = exponents[b * NPTS + n] * cen;
    float it  = intensities[b * NPTS + n] * cin;

    float ptx = points[(size_t)(b3 * NPTS + n) * 2 + 0];
    float pty = points[(size_t)(b3 * NPTS + n) * 2 + 1];
    float px = fminf(fmaxf((ptx - 128.0f) / 128.0f, -1.0f), 1.0f);
    float py = fminf(fmaxf((pty - 128.0f) / 128.0f, -1.0f), 1.0f);

    W[0 * 768 + g] = px;
    W[1 * 768 + g] = py;
    W[2 * 768 + g] = px * px + py * py;
    W[3 * 768 + g] = 1.0f / (2.0f * sig * sig);
    W[4 * 768 + g] = e;
    W[5 * 768 + g] = it;
}

// ---------------------------------------------------------------------------
// Kernel 2: WMMA distance GEMM + pow chain.
// grid = 3072 blocks x 256 threads = 24576 waves; 1 wave = 16 pixels x 128 pts.
// blockIdx.x / 512 = b3 (all 8 waves of a block share one camera image).
// ---------------------------------------------------------------------------
__global__ __launch_bounds__(256) void render_kernel(
    const float* __restrict__ W, float* __restrict__ out)
{
    __shared__ float sh[6 * NPTS];   // px, py, pr2, inv2s2, e, inten for this b3

    const int b3  = blockIdx.x >> 9;           // 512 blocks per camera
    const int tid = threadIdx.x;

    // stage this camera's per-point params into LDS
    for (int i = tid; i < 6 * NPTS; i += 256) {
        int a = i >> 7, n = i & (NPTS - 1);
        sh[i] = W[a * 768 + b3 * NPTS + n];
    }
    __syncthreads();

    const float* Lpx  = sh;
    const float* Lpy  = sh + 128;
    const float* Lpr2 = sh + 256;
    const float* Lis2 = sh + 384;
    const float* Le   = sh + 512;
    const float* Lit  = sh + 640;

    const int wave    = tid >> 5;
    const int lane    = tid & 31;
    const int lanelow = lane & 15;
    const int hi      = lane >> 4;             // 0: lanes 0-15, 1: lanes 16-31

    const int unit = blockIdx.x * 8 + wave;    // global work unit
    const int t    = unit & (TILES_PER_CAM - 1);
    const int pix0 = t << 4;                   // first pixel of this 16-pixel tile
    const int col0 = pix0 & (IMG - 1);
    const int row  = pix0 >> 8;                // tile never crosses a row (256%16==0)

    const float step = 2.0f / 255.0f;          // linspace(-1,1,256) step
    const float gx = -1.0f + (float)(col0 + lanelow) * step;
    const float gy = -1.0f + (float)row * step;

    // B operand (4x16 f32): lanes 0-15 hold K=0,1; lanes 16-31 hold K=2,3.
    //   K0 = -2*gx, K1 = -2*gy, K2 = 1, K3 = gx^2+gy^2
    v2f bmat;
    bmat.x = hi ? 1.0f : (-2.0f * gx);
    bmat.y = hi ? (gx * gx + gy * gy) : (-2.0f * gy);

    float* blobs = out + MASK_ELEMS;
    const int pixoff = pix0 + lanelow;
    float vmax = 0.0f;
    const float LOG2E = 1.44269504088896340f;

    #pragma unroll
    for (int mt = 0; mt < 8; ++mt) {
        // A operand (16x4 f32): lanes 0-15: K=0,1 = px,py; lanes 16-31: K=2,3 = pr2,1
        const int ia = mt * 16 + lanelow;
        v2f amat;
        amat.x = hi ? Lpr2[ia] : Lpx[ia];
        amat.y = hi ? 1.0f     : Lpy[ia];

        v8f d = {};
        // dst[m][n] = px*(-2gx) + py*(-2gy) + (px^2+py^2) + (gx^2+gy^2)
        d = __builtin_amdgcn_wmma_f32_16x16x4_f32(
                /*neg_a=*/false, amat, /*neg_b=*/false, bmat,
                /*c_mod=*/(short)0, d, /*reuse_a=*/false, /*reuse_b=*/false);

        #pragma unroll
        for (int r = 0; r < 8; ++r) {
            // D VGPR r: lanes 0-15 -> point mt*16+r, lanes 16-31 -> +8
            const int m  = mt * 16 + r + hi * 8;
            const float s  = Lis2[m];
            const float e  = Le[m];
            const float it = Lit[m];
            // clamp: GEMM-expanded distance can go slightly negative near dst=0
            float tt = fmaxf(d[r] * s, 0.0f);
            // blob = exp(-(tt)^e) = 2^(-log2e * 2^(e*log2 tt)); tt=0 -> blob=1
            float u    = __builtin_amdgcn_exp2f(e * __builtin_amdgcn_logf(tt));
            float blob = __builtin_amdgcn_exp2f(-LOG2E * u);
            blobs[(((size_t)(b3 * NPTS + m)) << 16) + pixoff] = blob;
            vmax = fmaxf(vmax, blob * it);
        }
    }

    // mask = min(max over all 128 points, 1): combine lane halves then store
    float other = __shfl_xor(vmax, 16, 32);
    float mk = fminf(fmaxf(vmax, other), 1.0f);
    if (hi == 0)
        out[(size_t)b3 * PIX + pixoff] = mk;
}

// ---------------------------------------------------------------------------
extern "C" void kernel_launch(void* const* d_in, const int* in_sizes, int n_in,
                              void* d_out, int out_size, void* d_ws, size_t ws_size,
                              hipStream_t stream) {
    const float* points      = (const float*)d_in[0];
    const float* sigmas      = (const float*)d_in[1];
    const float* exponents   = (const float*)d_in[2];
    const float* intensities = (const float*)d_in[3];
    const float* cam_sig     = (const float*)d_in[4];
    const float* cam_exp     = (const float*)d_in[5];
    const float* cam_int     = (const float*)d_in[6];
    float* W = (float*)d_ws;              // 6*768 floats = 18432 bytes

    hipLaunchKernelGGL(prep_kernel, dim3(3), dim3(256), 0, stream,
                       points, sigmas, exponents, intensities,
                       cam_sig, cam_exp, cam_int, W);

    hipLaunchKernelGGL(render_kernel, dim3(NB3 * TILES_PER_CAM / 8), dim3(256), 0, stream,
                       W, (float*)d_out);
}